// GCMCGraphConv_3959959847142
// MI455X (gfx1250) — compile-verified
//
#include <hip/hip_runtime.h>
#include <hip/hip_bf16.h>

#define D_FEAT 128
#define ROW_F4 32               // 128 floats = 32 float4
#define ROW_BYTES 512           // 128 floats * 4B
#define NBUF 4                  // async ring buffers per wave (issue-ahead = 3)
#define WAVES_PER_BLOCK 8       // 256 threads / wave32
#define BLOCK 256
#define EDGE_BLOCKS 4096        // 32768 waves, ~20 edges per wave

typedef __attribute__((address_space(3))) float4 lds_float4;

// ---------------------------------------------------------------------------
// Edge gather/scatter (placed first so the disasm snippet shows it).
// One wave per edge iteration: the 512B feat row of src[e] is DMA'd
// global->LDS with GLOBAL_LOAD_ASYNC_TO_LDS_B128 (16B/lane * 32 lanes),
// through a 4-deep ring so ~3 gathers stay in flight per wave (tracked by
// ASYNCcnt, zero VGPR cost). feat (51MB) is L2-resident (192MB L2), so the
// gathers run at L2 bandwidth; the scatter uses native no-return
// GLOBAL_ATOMIC_ADD_F32 resolved at the L2 atomic units (forced via asm so
// it can never fall back to a CAS loop).
// ---------------------------------------------------------------------------
__global__ __launch_bounds__(BLOCK) void gcmc_edge_kernel(
    const float* __restrict__ feat,
    const float* __restrict__ cj,
    const int*   __restrict__ src,
    const int*   __restrict__ dst,
    float*       __restrict__ out,
    int n_edges) {
  // 8 waves * 4 buffers * 32 float4 = 16KB LDS (WGP has 320KB).
  __shared__ float4 smem[WAVES_PER_BLOCK * NBUF * ROW_F4];

  const int lane  = threadIdx.x & 31;          // wave32
  const int wave  = threadIdx.x >> 5;
  const int gwave = blockIdx.x * WAVES_PER_BLOCK + wave;
  const int nwav  = gridDim.x * WAVES_PER_BLOCK;

  const unsigned long long feat_base = (unsigned long long)(const void*)feat;
  const unsigned long long out_base  = (unsigned long long)(void*)out;

  // Issue one async 16B-per-lane copy (feat row src[e] -> LDS buffer bi) and
  // prefetch this edge's cj normalizer so it rides the pipeline too.
  // The AS3 pointer operand is the 32-bit LDS offset the instruction's VDST
  // wants, AND it escapes smem into the asm so the compiler keeps our
  // subsequent ds_load reads (memory clobber => asm may write through it).
  auto issue_edge = [&](int e, unsigned bi) -> float {
    int s = src[e];
    unsigned long long ga = feat_base
                          + (unsigned long long)(unsigned)s * (unsigned long long)ROW_BYTES
                          + (unsigned long long)(lane * 16);
    lds_float4* la = (lds_float4*)(void*)
        &smem[(wave * NBUF + (int)bi) * ROW_F4 + lane];
    asm volatile("global_load_async_to_lds_b128 %0, %1, off"
                 :: "v"(la), "v"(ga)
                 : "memory");
    return cj[s];
  };

  // Wait until the oldest unconsumed copy has landed: async loads complete
  // in order, so asynccnt <= k leaves only the k newest outstanding.
  auto wait_newest = [&](unsigned k) {
    switch (k) {
      case 0: asm volatile("s_wait_asynccnt 0x0" ::: "memory"); break;
      case 1: asm volatile("s_wait_asynccnt 0x1" ::: "memory"); break;
      case 2: asm volatile("s_wait_asynccnt 0x2" ::: "memory"); break;
      default: asm volatile("s_wait_asynccnt 0x3" ::: "memory"); break;
    }
  };

  int e = gwave;
  if (e >= n_edges) return;

  // Prologue: fill up to 3 buffers ahead.
  float cj0 = issue_edge(e, 0u), cj1 = 0.f, cj2 = 0.f;
  unsigned nf = 1;                              // unconsumed filled buffers
  if (e + nwav < n_edges)     { cj1 = issue_edge(e + nwav,     1u); nf = 2; }
  if (e + 2 * nwav < n_edges) { cj2 = issue_edge(e + 2 * nwav, 2u); nf = 3; }

  unsigned bi = 0;                              // ring index of current buffer
  for (; e < n_edges; e += nwav) {
    const int e3 = e + 3 * nwav;
    float cj_new = 0.f;
    unsigned issued = 0;
    if (e3 < n_edges) {                         // keep the ring full
      cj_new = issue_edge(e3, (bi + 3u) & (NBUF - 1u));
      issued = 1;
    }

    const int d = dst[e];                       // overlap with the wait

    wait_newest(nf + issued - 1u);              // oldest (current) buffer ready

    // Typed LDS read -> ds_load_b128.
    const float4 v = smem[(wave * NBUF + (int)bi) * ROW_F4 + lane];

    // Native no-return f32 atomics (STOREcnt path, L2 atomic units).
    // S_ENDPGM does an implicit wait-idle, so these drain at kernel end.
    unsigned long long oa = out_base
                          + (unsigned long long)(unsigned)d * (unsigned long long)ROW_BYTES
                          + (unsigned long long)(lane * 16);
    asm volatile("global_atomic_add_f32 %0, %1, off"
                 :: "v"(oa), "v"(v.x * cj0) : "memory");
    asm volatile("global_atomic_add_f32 %0, %1, off offset:4"
                 :: "v"(oa), "v"(v.y * cj0) : "memory");
    asm volatile("global_atomic_add_f32 %0, %1, off offset:8"
                 :: "v"(oa), "v"(v.z * cj0) : "memory");
    asm volatile("global_atomic_add_f32 %0, %1, off offset:12"
                 :: "v"(oa), "v"(v.w * cj0) : "memory");

    cj0 = cj1; cj1 = cj2; cj2 = cj_new;         // rotate normalizer queue
    nf  = nf + issued - 1u;
    bi  = (bi + 1u) & (NBUF - 1u);
  }
}

// ---------------------------------------------------------------------------
// Zero the accumulator (d_out): harness poisons d_out and the edge kernel
// accumulates with atomics, so clear it every launch.
// ---------------------------------------------------------------------------
__global__ void gcmc_zero_kernel(float4* __restrict__ out, int n4) {
  int i = blockIdx.x * blockDim.x + threadIdx.x;
  if (i < n4) out[i] = make_float4(0.f, 0.f, 0.f, 0.f);
}

// ---------------------------------------------------------------------------
// out[row, :] *= ci[row], float4-vectorized (32 float4 per row).
// ---------------------------------------------------------------------------
__global__ void gcmc_scale_kernel(float4* __restrict__ out,
                                  const float* __restrict__ ci, int n4) {
  int i = blockIdx.x * blockDim.x + threadIdx.x;
  if (i < n4) {
    const float c = ci[i >> 5];                 // row = (i*4)/128 = i/32
    float4 v = out[i];
    v.x *= c; v.y *= c; v.z *= c; v.w *= c;
    out[i] = v;
  }
}

// ---------------------------------------------------------------------------
// Launch: inputs in setup_inputs() order: feat, cj, ci, src, dst.
// ---------------------------------------------------------------------------
extern "C" void kernel_launch(void* const* d_in, const int* in_sizes, int n_in,
                              void* d_out, int out_size, void* d_ws, size_t ws_size,
                              hipStream_t stream) {
  (void)n_in; (void)d_ws; (void)ws_size;

  const float* feat = (const float*)d_in[0];
  const float* cj   = (const float*)d_in[1];
  const float* ci   = (const float*)d_in[2];
  const int*   src  = (const int*)d_in[3];
  const int*   dst  = (const int*)d_in[4];
  float*       out  = (float*)d_out;

  const int n_edges = in_sizes[3];
  const int n4      = out_size / 4;             // float4 count (out_size % 4 == 0)

  gcmc_zero_kernel<<<(n4 + BLOCK - 1) / BLOCK, BLOCK, 0, stream>>>(
      (float4*)out, n4);

  gcmc_edge_kernel<<<EDGE_BLOCKS, BLOCK, 0, stream>>>(
      feat, cj, src, dst, out, n_edges);

  gcmc_scale_kernel<<<(n4 + BLOCK - 1) / BLOCK, BLOCK, 0, stream>>>(
      (float4*)out, ci, n4);
}